// Network_33792802685826
// MI455X (gfx1250) — compile-verified
//
#include <hip/hip_runtime.h>
#include <hip/hip_bf16.h>

// ---------------------------------------------------------------------------
// GCN forward on MI455X (gfx1250, wave32).
//  - Dense GEMMs: V_WMMA_F32_16X16X4_F32, one wave computes a 16x64 strip
//    (4 accumulators, A fragment reused across 4 WMMAs per k-step).
//  - Edge aggregation: per-block LDS staging of 256 edges (src/dst/norm read
//    once per edge), global_prefetch of gather rows, then 64 iterations of
//    4 edges x 64 features with ds-broadcast reads + global_atomic_add_f32.
//    All feature maps (25.6MB) are L2-resident (192MB L2).
// ---------------------------------------------------------------------------

typedef __attribute__((ext_vector_type(2))) float v2f;
typedef __attribute__((ext_vector_type(8))) float v8f;

#define HID 64
#define EPB 256   // edges staged per block in the scatter kernel

// ---------------- degree kernels ----------------
__global__ void k_deg_init(float* deg, int n) {
    int i = blockIdx.x * blockDim.x + threadIdx.x;
    if (i < n) deg[i] = 1.0f;   // self-loop
}

__global__ void k_deg_count(const long long* __restrict__ dst, float* deg, int E) {
    int i = blockIdx.x * blockDim.x + threadIdx.x;
    if (i < E) atomicAdd(&deg[(int)dst[i]], 1.0f);
}

__global__ void k_deg_rsqrt(float* deg, int n) {
    int i = blockIdx.x * blockDim.x + threadIdx.x;
    if (i < n) deg[i] = rsqrtf(deg[i]);
}

// ---------------- WMMA GEMM: H[M x 64] = A[M x K] * W[K x 64] ----------------
// Block = 256 threads = 8 waves. Each wave computes a 16x64 output strip:
// one A fragment per k-step feeds 4 WMMAs (one per 16-col tile).
template <int K>
__global__ void k_gemm_wmma(const float* __restrict__ A,
                            const float* __restrict__ W,
                            float* __restrict__ H, int M) {
    const int lane = threadIdx.x & 31;
    const int wave = threadIdx.x >> 5;
    const int row0 = (blockIdx.x * 8 + wave) * 16;
    if (row0 >= M) return;                                 // wave-uniform guard

    const int r     = row0 + (lane & 15);
    const int kHalf = (lane >> 4) * 2;   // lanes 0-15 -> K{0,1}; 16-31 -> K{2,3}
    const int c0    = lane & 15;

    v8f acc[4] = {};
    for (int k = 0; k < K; k += 4) {
        // A fragment: 2 consecutive K values for this lane's row (8B aligned)
        const float2 av = *(const float2*)(A + (size_t)r * K + (k + kHalf));
        v2f a; a.x = av.x; a.y = av.y;
        const float* Wp = W + (size_t)(k + kHalf) * HID + c0;
#pragma unroll
        for (int t = 0; t < 4; ++t) {
            v2f b;
            b.x = Wp[t * 16];
            b.y = Wp[t * 16 + HID];
            acc[t] = __builtin_amdgcn_wmma_f32_16x16x4_f32(
                         /*neg_a=*/false, a, /*neg_b=*/false, b,
                         /*c_mod=*/(short)0, acc[t],
                         /*reuse_a=*/false, /*reuse_b=*/false);
        }
    }

    // D store: VGPR i holds rows (row0+i) [lanes 0-15] and (row0+i+8) [16-31]
    const int rofs = (lane >> 4) * 8;
#pragma unroll
    for (int t = 0; t < 4; ++t) {
        float* out = H + (size_t)row0 * HID + t * 16 + c0;
#pragma unroll
        for (int i = 0; i < 8; ++i)
            out[(size_t)(i + rofs) * HID] = acc[t][i];
    }
}

// ---------------- aggregation ----------------
// agg[i,f] = h[i,f] * dis[i]^2   (self-loop term)
__global__ void k_agg_init(const float* __restrict__ h,
                           const float* __restrict__ dis,
                           float* __restrict__ agg, int nf) {
    int i = blockIdx.x * blockDim.x + threadIdx.x;
    if (i < nf) {
        float d = dis[i >> 6];
        agg[i] = h[i] * d * d;
    }
}

// agg[dst,:] += h[src,:] * dis[src]*dis[dst] over all edges.
// Phase 1: stage 256 edges (src, dst, norm) into LDS, one thread per edge,
//          and prefetch the gather rows into cache.
// Phase 2: 64 iterations x (4 edges x 64 features) with LDS broadcast reads.
__global__ void k_edge_scatter(const long long* __restrict__ src,
                               const long long* __restrict__ dst,
                               const float* __restrict__ dis,
                               const float* __restrict__ h,
                               float* __restrict__ agg, int E) {
    __shared__ int   ls[EPB];
    __shared__ int   ld[EPB];
    __shared__ float lw[EPB];

    const int tid = threadIdx.x;
    const int e0  = blockIdx.x * EPB;
    const int ei  = e0 + tid;
    if (ei < E) {
        const int s = (int)src[ei];
        const int d = (int)dst[ei];
        ls[tid] = s;
        ld[tid] = d;
        lw[tid] = dis[s] * dis[d];
        // warm the gather row (64 floats = 256B) for phase 2
        __builtin_prefetch(h + (size_t)s * HID, 0, 0);
        __builtin_prefetch(h + (size_t)s * HID + 32, 0, 0);
    }
    __syncthreads();

    const int f   = tid & 63;   // feature
    const int sub = tid >> 6;   // which edge within each group of 4
    int limit = E - e0; if (limit > EPB) limit = EPB;

    if (limit == EPB) {
#pragma unroll 8
        for (int j = sub; j < EPB; j += 4) {
            const int s = ls[j], d = ld[j];
            const float w = lw[j];
            atomicAdd(&agg[(size_t)d * HID + f], h[(size_t)s * HID + f] * w);
        }
    } else {
        for (int j = sub; j < limit; j += 4) {
            const int s = ls[j], d = ld[j];
            const float w = lw[j];
            atomicAdd(&agg[(size_t)d * HID + f], h[(size_t)s * HID + f] * w);
        }
    }
}

// out = (relu?) (agg + bias)
__global__ void k_finalize(float* __restrict__ agg,
                           const float* __restrict__ bias,
                           int nf, int do_relu) {
    int i = blockIdx.x * blockDim.x + threadIdx.x;
    if (i < nf) {
        float v = agg[i] + bias[i & 63];
        agg[i] = do_relu ? fmaxf(v, 0.0f) : v;
    }
}

// ---------------- pooling ----------------
__global__ void k_zero(float* p, int n) {
    int i = blockIdx.x * blockDim.x + threadIdx.x;
    if (i < n) p[i] = 0.0f;
}

__global__ void k_pool(const float* __restrict__ h,
                       const long long* __restrict__ b,
                       float* __restrict__ g, int nf) {
    int i = blockIdx.x * blockDim.x + threadIdx.x;
    if (i < nf) {
        int graph = (int)b[i >> 6];
        atomicAdd(&g[(size_t)graph * HID + (i & 63)], h[i]);
    }
}

// ---------------- MLP head: out = relu(g@Wl1+bl1)@Wl2+bl2 ----------------
__global__ void k_mlp(const float* __restrict__ g,
                      const float* __restrict__ Wl1, const float* __restrict__ bl1,
                      const float* __restrict__ Wl2, const float* __restrict__ bl2,
                      float* __restrict__ out, int G) {
    int i = blockIdx.x * blockDim.x + threadIdx.x;
    if (i >= G) return;
    float gr[HID];
#pragma unroll
    for (int k = 0; k < HID; ++k) gr[k] = g[(size_t)i * HID + k];
    float acc = bl2[0];
#pragma unroll 4
    for (int j = 0; j < 32; ++j) {
        float hj = bl1[j];
#pragma unroll
        for (int k = 0; k < HID; ++k) hj = fmaf(gr[k], Wl1[k * 32 + j], hj);
        acc = fmaf(fmaxf(hj, 0.0f), Wl2[j], acc);
    }
    out[i] = acc;
}

// ---------------------------------------------------------------------------
static inline int cdiv(long long a, long long b) { return (int)((a + b - 1) / b); }

extern "C" void kernel_launch(void* const* d_in, const int* in_sizes, int n_in,
                              void* d_out, int out_size, void* d_ws, size_t ws_size,
                              hipStream_t stream) {
    const float*      x   = (const float*)d_in[0];
    const long long*  e   = (const long long*)d_in[1];
    const long long*  b   = (const long long*)d_in[2];
    const float*      W1  = (const float*)d_in[3];
    const float*      b1  = (const float*)d_in[4];
    const float*      W2  = (const float*)d_in[5];
    const float*      b2  = (const float*)d_in[6];
    const float*      W3  = (const float*)d_in[7];
    const float*      b3  = (const float*)d_in[8];
    const float*      Wl1 = (const float*)d_in[9];
    const float*      bl1 = (const float*)d_in[10];
    const float*      Wl2 = (const float*)d_in[11];
    const float*      bl2 = (const float*)d_in[12];

    const int N = in_sizes[2];            // 100000 nodes
    const int E = in_sizes[1] / 2;        // 1.6M edges
    const int G = out_size;               // 2048 graphs
    const long long* src = e;
    const long long* dst = e + E;

    // workspace layout (bytes): dis[N] | hA[N*64] | hB[N*64] | g[G*64]
    char*  ws  = (char*)d_ws;
    float* dis = (float*)ws;
    size_t off = ((size_t)N * 4 + 255) & ~(size_t)255;
    float* hA  = (float*)(ws + off);
    float* hB  = hA + (size_t)N * HID;
    float* gp  = hB + (size_t)N * HID;
    (void)ws_size; (void)n_in;

    const int B  = 256;
    const int nf = N * HID;          // 6.4M
    const int gemmGrid = cdiv(N, 128);   // 8 waves x 16 rows per block

    // degrees: deg -> rsqrt(deg) in place
    k_deg_init <<<cdiv(N, B), B, 0, stream>>>(dis, N);
    k_deg_count<<<cdiv(E, B), B, 0, stream>>>(dst, dis, E);
    k_deg_rsqrt<<<cdiv(N, B), B, 0, stream>>>(dis, N);

    // ---- layer 1: x[N,128] @ W1 -> hA; aggregate -> hB; relu ----
    k_gemm_wmma<128><<<gemmGrid, B, 0, stream>>>(x, W1, hA, N);
    k_agg_init     <<<cdiv(nf, B), B, 0, stream>>>(hA, dis, hB, nf);
    k_edge_scatter <<<cdiv(E, EPB), B, 0, stream>>>(src, dst, dis, hA, hB, E);
    k_finalize     <<<cdiv(nf, B), B, 0, stream>>>(hB, b1, nf, 1);

    // ---- layer 2: hB @ W2 -> hA; aggregate -> hB; relu ----
    k_gemm_wmma<64><<<gemmGrid, B, 0, stream>>>(hB, W2, hA, N);
    k_agg_init     <<<cdiv(nf, B), B, 0, stream>>>(hA, dis, hB, nf);
    k_edge_scatter <<<cdiv(E, EPB), B, 0, stream>>>(src, dst, dis, hA, hB, E);
    k_finalize     <<<cdiv(nf, B), B, 0, stream>>>(hB, b2, nf, 1);

    // ---- layer 3: hB @ W3 -> hA; aggregate -> hB; no relu ----
    k_gemm_wmma<64><<<gemmGrid, B, 0, stream>>>(hB, W3, hA, N);
    k_agg_init     <<<cdiv(nf, B), B, 0, stream>>>(hA, dis, hB, nf);
    k_edge_scatter <<<cdiv(E, EPB), B, 0, stream>>>(src, dst, dis, hA, hB, E);
    k_finalize     <<<cdiv(nf, B), B, 0, stream>>>(hB, b3, nf, 0);

    // ---- pool + MLP head ----
    k_zero<<<cdiv(G * HID, B), B, 0, stream>>>(gp, G * HID);
    k_pool<<<cdiv(nf, B), B, 0, stream>>>(hB, b, gp, nf);
    k_mlp <<<cdiv(G, B), B, 0, stream>>>(gp, Wl1, bl1, Wl2, bl2, (float*)d_out, G);
}